// ConductivityPredictor_1829656068195
// MI455X (gfx1250) — compile-verified
//
#include <hip/hip_runtime.h>
#include <hip/hip_bf16.h>
#include <math.h>

// ---------------------------------------------------------------------------
// Problem constants (fixed by the reference)
// ---------------------------------------------------------------------------
#define NN 50000      // nodes
#define NE 150000     // edges
#define NG 256        // graphs
#define CD 512        // channels
#define FIN 118       // input features
#define KPAD 120      // FIN padded to a multiple of 4
#define NL 4          // layers

typedef float v2f __attribute__((ext_vector_type(2)));
typedef float v8f __attribute__((ext_vector_type(8)));

// ---------------------------------------------------------------------------
// fp32 WMMA GEMM:  out[M,N] = act(A[M,K] @ W[K,N] + bias[N])
// W is PRE-PACKED as Wp[k>>2][n][k&3] so each lane's B pair is one b64 load.
// Block = 256 threads = 8 waves. Each wave -> 16 (M) x 128 (N) tile
// (8 accumulators). Block tile = 128 x 128. grid = (N/128, ceil(M/128)).
// V_WMMA_F32_16X16X4_F32 (exact fp32, K-step 4). Branchless inner loop with
// register double-buffering so WMMAs overlap the next step's loads.
// ---------------------------------------------------------------------------
__global__ __launch_bounds__(256) void gemm_bias_act_kernel(
    const float* __restrict__ A, const float* __restrict__ Wp,
    const float* __restrict__ bias, float* __restrict__ out,
    int M, int N, int K, int act)
{
    const int lane = threadIdx.x & 31;
    const int wv   = threadIdx.x >> 5;
    const int row0 = blockIdx.y * 128 + wv * 16;
    const int n0   = blockIdx.x * 128;

    // A fragment (32-bit A 16x4 layout):
    //   lanes 0-15 : M=lane,    K = k0 + {0,1}
    //   lanes16-31 : M=lane-16, K = k0 + {2,3}
    const int koff = (lane >> 4) * 2;
    int aRow = row0 + (lane & 15);
    aRow = aRow < M ? aRow : (M - 1);                 // clamp: no branches in loop
    const float* ap = A + (size_t)aRow * K + koff;    // +4 per k-step

    // B fragment from packed layout: Wp[((k>>2)*N + n)*4 + (k&3)]
    //   lane reads b64 at ((s*N + col)*4 + koff); subtile t offset = t*64 floats
    const float* bp = Wp + (size_t)(n0 + (lane & 15)) * 4 + koff;  // +4*N per step

    v8f acc[8] = {};   // eight 16x16 fp32 accumulators across N

    const int steps  = K >> 2;
    const size_t bstride = (size_t)N * 4;

    // prologue: fragments for step 0
    v2f a_cur = *(const v2f*)ap;
    v2f b_cur[8];
#pragma unroll
    for (int t = 0; t < 8; ++t) b_cur[t] = *(const v2f*)(bp + t * 64);

#pragma unroll 2
    for (int s = 0; s < steps - 1; ++s) {
        ap += 4;
        bp += bstride;
        // issue next step's loads before this step's WMMAs
        v2f a_nxt = *(const v2f*)ap;
        v2f b_nxt[8];
#pragma unroll
        for (int t = 0; t < 8; ++t) b_nxt[t] = *(const v2f*)(bp + t * 64);
#pragma unroll
        for (int t = 0; t < 8; ++t)
            acc[t] = __builtin_amdgcn_wmma_f32_16x16x4_f32(
                false, a_cur, false, b_cur[t], (short)0, acc[t], false, false);
        a_cur = a_nxt;
#pragma unroll
        for (int t = 0; t < 8; ++t) b_cur[t] = b_nxt[t];
    }
    // final step
#pragma unroll
    for (int t = 0; t < 8; ++t)
        acc[t] = __builtin_amdgcn_wmma_f32_16x16x4_f32(
            false, a_cur, false, b_cur[t], (short)0, acc[t], false, false);

    // Epilogue. C/D 16x16 layout: VGPR r -> M=r (lanes 0-15), M=r+8 (lanes 16-31).
    const int rowHi = (lane >> 4) * 8;
#pragma unroll
    for (int t = 0; t < 8; ++t) {
        const int col = n0 + t * 16 + (lane & 15);
        const float bv = bias[col];
#pragma unroll
        for (int r = 0; r < 8; ++r) {
            const int orow = row0 + rowHi + r;
            if (orow < M) {
                float v = acc[t][r] + bv;
                if (act) // exact-erf GELU (matches jax approximate=False)
                    v = 0.5f * v * (1.0f + erff(v * 0.70710678118654752440f));
                out[(size_t)orow * N + col] = v;
            }
        }
    }
}

// ---------------------------------------------------------------------------
// Weight packing: Wp[((k>>2)*CD + n)*4 + (k&3)] = (k < Ksrc) ? W[k*CD+n] : 0
// (zero-fill handles the embed K=118 -> 120 padding)
// ---------------------------------------------------------------------------
__global__ void pack_w_kernel(const float* __restrict__ W, float* __restrict__ Wp,
                              int K, int Ksrc) {
    int idx = blockIdx.x * blockDim.x + threadIdx.x;
    if (idx >= K * CD) return;
    int k = idx >> 9;          // / CD
    int n = idx & (CD - 1);    // % CD
    float v = (k < Ksrc) ? W[(size_t)k * CD + n] : 0.0f;
    Wp[((size_t)(k >> 2) * CD + n) * 4 + (k & 3)] = v;
}

// x padded to [NN, KPAD] with zero tail
__global__ void pad_x_kernel(const float* __restrict__ x, float* __restrict__ xpad) {
    int idx = blockIdx.x * blockDim.x + threadIdx.x;
    if (idx >= NN * KPAD) return;
    int i = idx / KPAD;
    int c = idx - i * KPAD;
    xpad[idx] = (c < FIN) ? x[(size_t)i * FIN + c] : 0.0f;
}

// ---------------------------------------------------------------------------
// Utility / graph kernels
// ---------------------------------------------------------------------------
__global__ void zero_kernel(float* __restrict__ p, int n) {
    int i = blockIdx.x * blockDim.x + threadIdx.x;
    if (i < n) p[i] = 0.0f;
}

__global__ void deg_count_kernel(const long long* __restrict__ eidx,
                                 float* __restrict__ deg) {
    int e = blockIdx.x * blockDim.x + threadIdx.x;
    if (e < NE) atomicAdd(&deg[(int)eidx[NE + e]], 1.0f);   // dst row
}

__global__ void inv_deg_kernel(float* __restrict__ deg) {
    int i = blockIdx.x * blockDim.x + threadIdx.x;
    if (i < NN) deg[i] = 1.0f / fmaxf(deg[i], 1.0f);
}

__global__ void batch_count_kernel(const long long* __restrict__ batch,
                                   float* __restrict__ gcnt) {
    int i = blockIdx.x * blockDim.x + threadIdx.x;
    if (i < NN) atomicAdd(&gcnt[(int)batch[i]], 1.0f);
}

// agg[dst, :] += msg[src, :] over all edges (one thread per edge-channel)
__global__ void edge_scatter_kernel(const float* __restrict__ msg,
                                    const long long* __restrict__ eidx,
                                    float* __restrict__ agg) {
    int idx = blockIdx.x * blockDim.x + threadIdx.x;
    if (idx >= NE * CD) return;
    int e = idx >> 9;          // / CD
    int c = idx & (CD - 1);    // % CD
    int s = (int)eidx[e];
    int d = (int)eidx[NE + e];
    atomicAdd(&agg[(size_t)d * CD + c], msg[(size_t)s * CD + c]);
}

// agg[i, :] *= inv_deg[i]
__global__ void row_scale_kernel(float* __restrict__ agg,
                                 const float* __restrict__ invd) {
    int idx = blockIdx.x * blockDim.x + threadIdx.x;
    if (idx < NN * CD) agg[idx] *= invd[idx >> 9];
}

// pooled[batch[i], :] += h[i, :]
__global__ void pool_scatter_kernel(const float* __restrict__ h,
                                    const long long* __restrict__ batch,
                                    float* __restrict__ pooled) {
    int idx = blockIdx.x * blockDim.x + threadIdx.x;
    if (idx >= NN * CD) return;
    int i = idx >> 9;
    int c = idx & (CD - 1);
    atomicAdd(&pooled[(size_t)((int)batch[i]) * CD + c], h[idx]);
}

// out[g] = dot(pooled[g,:], head_W) / max(gcnt[g],1) + head_b
__global__ __launch_bounds__(256) void head_kernel(
    const float* __restrict__ pooled, const float* __restrict__ gcnt,
    const float* __restrict__ headW, const float* __restrict__ headB,
    float* __restrict__ out)
{
    __shared__ float sred[256];
    const int g = blockIdx.x;
    float p = 0.0f;
    for (int c = threadIdx.x; c < CD; c += 256)
        p += pooled[(size_t)g * CD + c] * headW[c];
    sred[threadIdx.x] = p;
    __syncthreads();
    for (int s = 128; s > 0; s >>= 1) {
        if (threadIdx.x < s) sred[threadIdx.x] += sred[threadIdx.x + s];
        __syncthreads();
    }
    if (threadIdx.x == 0)
        out[g] = sred[0] / fmaxf(gcnt[g], 1.0f) + headB[0];
}

// ---------------------------------------------------------------------------
// Host orchestration
// ---------------------------------------------------------------------------
extern "C" void kernel_launch(void* const* d_in, const int* in_sizes, int n_in,
                              void* d_out, int out_size, void* d_ws, size_t ws_size,
                              hipStream_t stream) {
    (void)in_sizes; (void)n_in; (void)out_size; (void)ws_size;

    const float*     x      = (const float*)d_in[0];
    const long long* eidx   = (const long long*)d_in[1];   // int64 [2, NE]
    const long long* batch  = (const long long*)d_in[2];   // int64 [NN]
    const float*     embW   = (const float*)d_in[3];
    const float*     embB   = (const float*)d_in[4];
    const float*     W1     = (const float*)d_in[5];
    const float*     b1     = (const float*)d_in[6];
    const float*     W2     = (const float*)d_in[7];
    const float*     b2     = (const float*)d_in[8];
    const float*     headW  = (const float*)d_in[9];
    const float*     headB  = (const float*)d_in[10];
    float*           out    = (float*)d_out;

    // Workspace carve (256B aligned)
    char*  base = (char*)d_ws;
    size_t off  = 0;
    auto carve = [&](size_t bytes) -> float* {
        float* p = (float*)(base + off);
        off = (off + bytes + 255) & ~(size_t)255;
        return p;
    };
    float* bufA    = carve((size_t)NN * CD * sizeof(float));     // ping
    float* bufB    = carve((size_t)NN * CD * sizeof(float));     // pong
    float* xpad    = carve((size_t)NN * KPAD * sizeof(float));   // x padded (K=120)
    float* wpEmb   = carve((size_t)KPAD * CD * sizeof(float));   // packed embed W
    float* wp1     = carve((size_t)NL * CD * CD * sizeof(float));// packed W1[l]
    float* wp2     = carve((size_t)NL * CD * CD * sizeof(float));// packed W2[l]
    float* deg     = carve((size_t)NN * sizeof(float));          // inverse in-degree
    float* gcnt    = carve((size_t)NG * sizeof(float));          // nodes per graph
    float* pooled  = carve((size_t)NG * CD * sizeof(float));

    const int T = 256;
    const dim3 gemmGrid(CD / 128, (NN + 127) / 128);
    const int nodeChanBlocks = (NN * CD + T - 1) / T;
    const int edgeChanBlocks = (NE * CD + T - 1) / T;

    // Degree counts and per-graph counts (constant across layers)
    zero_kernel<<<(NN + T - 1) / T, T, 0, stream>>>(deg, NN);
    zero_kernel<<<(NG + T - 1) / T, T, 0, stream>>>(gcnt, NG);
    deg_count_kernel<<<(NE + T - 1) / T, T, 0, stream>>>(eidx, deg);
    batch_count_kernel<<<(NN + T - 1) / T, T, 0, stream>>>(batch, gcnt);
    inv_deg_kernel<<<(NN + T - 1) / T, T, 0, stream>>>(deg);

    // Pad x; pack all weight matrices into the WMMA-fragment-friendly layout
    pad_x_kernel<<<(NN * KPAD + T - 1) / T, T, 0, stream>>>(x, xpad);
    pack_w_kernel<<<(KPAD * CD + T - 1) / T, T, 0, stream>>>(embW, wpEmb, KPAD, FIN);
    for (int l = 0; l < NL; ++l) {
        pack_w_kernel<<<(CD * CD + T - 1) / T, T, 0, stream>>>(
            W1 + (size_t)l * CD * CD, wp1 + (size_t)l * CD * CD, CD, CD);
        pack_w_kernel<<<(CD * CD + T - 1) / T, T, 0, stream>>>(
            W2 + (size_t)l * CD * CD, wp2 + (size_t)l * CD * CD, CD, CD);
    }

    // Embed: h = x @ embW + embB   (K = 120, zero-padded)
    gemm_bias_act_kernel<<<gemmGrid, T, 0, stream>>>(
        xpad, wpEmb, embB, bufA, NN, CD, KPAD, /*act=*/0);

    float* h     = bufA;
    float* other = bufB;
    for (int l = 0; l < NL; ++l) {
        // msg = gelu(h @ W1[l] + b1[l])
        gemm_bias_act_kernel<<<gemmGrid, T, 0, stream>>>(
            h, wp1 + (size_t)l * CD * CD, b1 + (size_t)l * CD, other,
            NN, CD, CD, /*act=*/1);
        // agg = scatter_mean(msg[src], dst)  (reuse h's buffer for agg)
        zero_kernel<<<nodeChanBlocks, T, 0, stream>>>(h, NN * CD);
        edge_scatter_kernel<<<edgeChanBlocks, T, 0, stream>>>(other, eidx, h);
        row_scale_kernel<<<nodeChanBlocks, T, 0, stream>>>(h, deg);
        // h = gelu(agg @ W2[l] + b2[l])
        gemm_bias_act_kernel<<<gemmGrid, T, 0, stream>>>(
            h, wp2 + (size_t)l * CD * CD, b2 + (size_t)l * CD, other,
            NN, CD, CD, /*act=*/1);
        float* tmp = h; h = other; other = tmp;
    }

    // Global mean pool + head
    zero_kernel<<<(NG * CD + T - 1) / T, T, 0, stream>>>(pooled, NG * CD);
    pool_scatter_kernel<<<nodeChanBlocks, T, 0, stream>>>(h, batch, pooled);
    head_kernel<<<NG, T, 0, stream>>>(pooled, gcnt, headW, headB, out);
}